// CostSensitiveCrossEntropyLossN__57037165690983
// MI455X (gfx1250) — compile-verified
//
#include <hip/hip_runtime.h>
#include <hip/hip_bf16.h>
#include <math.h>

// ---------------------------------------------------------------------------
// CostSensitiveCrossEntropyLoss for MI455X (gfx1250, wave32).
// Bandwidth-bound: 1.05 GB logits read once => ~45us floor @ 23.3 TB/s.
// One wave32 per row, b128 streaming loads, row kept in registers, WMMA-based
// wave reductions (v_wmma_f32_16x16x4_f32 vs ones B: D[m][n] = v[m]+v[m+16]).
// ---------------------------------------------------------------------------

#define SCALE_CONST 1.0f

typedef __attribute__((ext_vector_type(2)))  float v2f;
typedef __attribute__((ext_vector_type(8)))  float v8f;

// Wave32 sum reduction via WMMA. Requires full EXEC (call only from
// wave-uniform control flow). Returns the full 32-lane sum in every lane.
__device__ __forceinline__ float wave_sum_wmma(float v) {
  // A (16x4 f32, 2 VGPRs): VGPR0 = v (K=0 for lanes 0-15, K=2 for lanes 16-31),
  // VGPR1 = 0. B = ones(4x16). D[m][n] = v[m] + v[m+16] for every n.
  v2f a; a[0] = v;    a[1] = 0.0f;
  v2f b; b[0] = 1.0f; b[1] = 1.0f;
  v8f c = {};
  c = __builtin_amdgcn_wmma_f32_16x16x4_f32(
      /*neg_a=*/false, a, /*neg_b=*/false, b,
      /*c_mod=*/(short)0, c, /*reuse_a=*/false, /*reuse_b=*/false);
  // Lanes 0-15 hold rows 0..7 of D in c[0..7]; lanes 16-31 hold rows 8..15.
  float half_sum = c[0] + c[1] + c[2] + c[3] + c[4] + c[5] + c[6] + c[7];
  // Cross-half combine: total = sum over all 16 rows = sum of all 32 v's.
  return half_sum + __shfl_xor(half_sum, 16, 32);
}

// ---------------------------------------------------------------------------
// K0: init cm from the (zero) input cost matrix; zero maxbits and the output.
__global__ void cscel_init_kernel(const float* __restrict__ cm_in,
                                  float* __restrict__ cm,
                                  unsigned* __restrict__ maxbits,
                                  float* __restrict__ out, int n) {
  int stride = gridDim.x * blockDim.x;
  for (int i = blockIdx.x * blockDim.x + threadIdx.x; i < n; i += stride)
    cm[i] = cm_in[i];
  if (blockIdx.x == 0 && threadIdx.x == 0) {
    *maxbits = 0u;   // 0.0f
    out[0] = 0.0f;
  }
}

// ---------------------------------------------------------------------------
// K1: one wave32 per row (supports C <= 1024). Streams the row once with
// b128 loads (512B per wave per instruction), keeps it in registers, computes
// max+argmax (first-index tiebreak), WMMA-reduced sum(exp(x-max)), writes
// pred / gathered log-prob, scatter-adds the confusion-matrix count.
// OOB slots hold -inf so exp(x-m) contributes exactly 0 without guards.
template <bool VEC4>
__global__ void __launch_bounds__(256)
cscel_row_kernel(const float* __restrict__ logits,
                 const int* __restrict__ targets,
                 int* __restrict__ pred,
                 float* __restrict__ glp,
                 float* __restrict__ cm,
                 int N, int C) {
  const int lane = threadIdx.x & 31;
  const int wave = (blockIdx.x * blockDim.x + threadIdx.x) >> 5;  // row id
  if (wave >= N) return;  // wave-uniform: EXEC stays all-ones below

  const float* __restrict__ row = logits + (size_t)wave * (size_t)C;
  const float NEG_INF = -__builtin_inff();

  float m = NEG_INF;
  int mi = C;  // sentinel larger than any index
  float s;     // per-lane partial of sum(exp(x - m)), after max is known

  if constexpr (VEC4) {
    // ---- vector pass: 8 float4 per lane covers C <= 1024 ----
    const int NV4 = C >> 2;
    const float4* __restrict__ row4 = reinterpret_cast<const float4*>(row);
    float4 vals[8];
#pragma unroll
    for (int k = 0; k < 8; ++k) {
      int v4 = lane + 32 * k;
      float4 x;
      if (v4 < NV4) x = row4[v4];
      else { x.x = NEG_INF; x.y = NEG_INF; x.z = NEG_INF; x.w = NEG_INF; }
      vals[k] = x;
      int base = 4 * v4;  // in-lane indices increase -> strict '>' keeps first
      if (x.x > m) { m = x.x; mi = base;     }
      if (x.y > m) { m = x.y; mi = base + 1; }
      if (x.z > m) { m = x.z; mi = base + 2; }
      if (x.w > m) { m = x.w; mi = base + 3; }
    }
    // wave32 max + argmax (lowest index on ties, like jnp.argmax)
#pragma unroll
    for (int off = 16; off > 0; off >>= 1) {
      float om = __shfl_xor(m, off, 32);
      int   oi = __shfl_xor(mi, off, 32);
      if (om > m || (om == m && oi < mi)) { m = om; mi = oi; }
    }
    // unguarded exp-sum: -inf slots contribute 0
    s = 0.0f;
#pragma unroll
    for (int k = 0; k < 8; ++k) {
      float4 x = vals[k];
      s += __expf(x.x - m);
      s += __expf(x.y - m);
      s += __expf(x.z - m);
      s += __expf(x.w - m);
    }
  } else {
    // ---- scalar pass: 32 floats per lane covers C <= 1024 ----
    float vals[32];
#pragma unroll
    for (int k = 0; k < 32; ++k) {
      int c = lane + 32 * k;
      float x = (c < C) ? row[c] : NEG_INF;
      vals[k] = x;
      if (x > m) { m = x; mi = c; }
    }
#pragma unroll
    for (int off = 16; off > 0; off >>= 1) {
      float om = __shfl_xor(m, off, 32);
      int   oi = __shfl_xor(mi, off, 32);
      if (om > m || (om == m && oi < mi)) { m = om; mi = oi; }
    }
    s = 0.0f;
#pragma unroll
    for (int k = 0; k < 32; ++k) s += __expf(vals[k] - m);
  }

  float tot = wave_sum_wmma(s);   // full EXEC here

  if (lane == 0) {
    int t = targets[wave];
    float xt = row[t];                          // L0/L2 hit: row just streamed
    glp[wave]  = (xt - m) - __logf(tot);        // log_softmax at target class
    pred[wave] = mi;
    atomicAdd(&cm[(size_t)t * C + mi], 1.0f);   // confusion-matrix count
  }
}

// ---------------------------------------------------------------------------
// K2: max over off-diagonal cm entries. Counts are >= 0 floats, so their bit
// patterns order like unsigned ints -> atomicMax on bits.
__global__ void cscel_cm_max_kernel(const float* __restrict__ cm,
                                    unsigned* __restrict__ maxbits, int C) {
  __shared__ unsigned smax;
  if (threadIdx.x == 0) smax = 0u;
  __syncthreads();

  unsigned local = 0u;
  int total = C * C;
  int stride = gridDim.x * blockDim.x;
  for (int i = blockIdx.x * blockDim.x + threadIdx.x; i < total; i += stride) {
    int r = i / C;
    int c = i - r * C;
    if (r != c) {
      unsigned b = __float_as_uint(cm[i]);
      local = (b > local) ? b : local;
    }
  }
  atomicMax(&smax, local);
  __syncthreads();
  if (threadIdx.x == 0) atomicMax(maxbits, smax);
}

// ---------------------------------------------------------------------------
// K3: loss = -mean(glp[n] * cost(t_n, p_n)),
//     cost(t,p) = (t!=p ? cm[t,p] : 0)/maxv * SCALE + 1.
// WMMA wave reduction, one atomicAdd per wave into out[0].
__global__ void __launch_bounds__(256)
cscel_loss_kernel(const int* __restrict__ targets,
                  const int* __restrict__ pred,
                  const float* __restrict__ glp,
                  const float* __restrict__ cm,
                  const unsigned* __restrict__ maxbits,
                  float* __restrict__ out,
                  int N, int C, float neg_inv_n) {
  float maxv = __uint_as_float(*maxbits);
  float inv_max_scale = SCALE_CONST / maxv;

  float acc = 0.0f;
  int stride = gridDim.x * blockDim.x;
  for (int i = blockIdx.x * blockDim.x + threadIdx.x; i < N; i += stride) {
    int t = targets[i];
    int p = pred[i];
    float cnt = (t != p) ? cm[(size_t)t * C + p] : 0.0f;
    float cost = cnt * inv_max_scale + 1.0f;
    acc += glp[i] * cost;
  }
  // Loop reconverges -> full EXEC for the WMMA reduction.
  float wsum = wave_sum_wmma(acc);
  if ((threadIdx.x & 31) == 0) atomicAdd(out, wsum * neg_inv_n);
}

// ---------------------------------------------------------------------------
extern "C" void kernel_launch(void* const* d_in, const int* in_sizes, int n_in,
                              void* d_out, int out_size, void* d_ws, size_t ws_size,
                              hipStream_t stream) {
  const float* logits   = (const float*)d_in[0];   // [N, C] f32
  const float* cm_in    = (const float*)d_in[1];   // [C, C] f32 (zeros)
  const int*   targets  = (const int*)d_in[2];     // [N] int

  const int CC = in_sizes[1];                      // C*C
  int C = (int)(sqrtf((float)CC) + 0.5f);          // 1000
  const int N = in_sizes[2];                       // 262144

  // Workspace layout: cm [C*C f32] | pred [N i32] | glp [N f32] | maxbits [u32]
  float*    cm      = (float*)d_ws;
  int*      pred    = (int*)(cm + (size_t)CC);
  float*    glp     = (float*)(pred + N);
  unsigned* maxbits = (unsigned*)(glp + N);
  float*    out     = (float*)d_out;

  // K0: init cm / maxbits / out
  cscel_init_kernel<<<256, 256, 0, stream>>>(cm_in, cm, maxbits, out, CC);

  // K1: one wave32 per row, 8 waves per 256-thread block
  const int wavesPerBlock = 256 / 32;
  int grid1 = (N + wavesPerBlock - 1) / wavesPerBlock;
  if ((C & 3) == 0) {
    cscel_row_kernel<true><<<grid1, 256, 0, stream>>>(logits, targets, pred,
                                                      glp, cm, N, C);
  } else {
    cscel_row_kernel<false><<<grid1, 256, 0, stream>>>(logits, targets, pred,
                                                       glp, cm, N, C);
  }

  // K2: off-diagonal max of the confusion matrix (4 MB, L2-resident)
  cscel_cm_max_kernel<<<512, 256, 0, stream>>>(cm, maxbits, C);

  // K3: final reduction into the scalar loss
  cscel_loss_kernel<<<512, 256, 0, stream>>>(targets, pred, glp, cm, maxbits,
                                             out, N, C, -1.0f / (float)N);
}